// GraphAttentionLayer_34376918237284
// MI455X (gfx1250) — compile-verified
//
#include <hip/hip_runtime.h>
#include <hip/hip_bf16.h>
#include <math.h>

#define N_NODES   50000
#define N_EDGES   800000
#define IN_F      128
#define OUT_F     32
#define HEADS     4
#define HF        128          // HEADS * OUT_F
#define NEG_SLOPE 0.2f
#define GAT_EPS   1e-8f

typedef float v2f __attribute__((ext_vector_type(2)));
typedef float v8f __attribute__((ext_vector_type(8)));

// ---------------------------------------------------------------------------
// K1: h = x @ W^T   via V_WMMA_F32_16X16X4_F32 (one 16x16 output tile / wave)
//   A(16x4)  = x rows,  element (m,k): lane = (k>=2)*16 + m, vgpr = k&1
//   B(4x16)  = W^T,     element (k,n): lane = (k>=2)*16 + n, vgpr = k&1
//   D(16x16): vgpr r: lanes 0-15 -> M=r, lanes 16-31 -> M=r+8 ; N = lane&15
// ---------------------------------------------------------------------------
__global__ __launch_bounds__(128) void gat_gemm_wmma(const float* __restrict__ X,
                                                     const float* __restrict__ W,
                                                     float* __restrict__ Hout) {
  const int lane = threadIdx.x & 31;
  const int wave = threadIdx.x >> 5;
  const int row0 = blockIdx.x * 16;                    // node tile
  const int col0 = (blockIdx.y * 4 + wave) * 16;       // out-feature tile
  const int mn = lane & 15;                            // M for A, N for B
  const int kb = (lane >> 4) * 2;                      // K sub-offset 0 or 2

  const float* __restrict__ xrow = X + (size_t)(row0 + mn) * IN_F + kb;
  const float* __restrict__ wrow = W + (size_t)(col0 + mn) * IN_F + kb;

  v8f acc = {};
#pragma unroll 8
  for (int k0 = 0; k0 < IN_F; k0 += 4) {
    v2f a = *(const v2f*)(xrow + k0);
    v2f b = *(const v2f*)(wrow + k0);
    acc = __builtin_amdgcn_wmma_f32_16x16x4_f32(false, a, false, b,
                                                (short)0, acc, false, false);
  }

  const int mo = (lane >> 4) * 8;
  float* __restrict__ dst = Hout + (size_t)(row0 + mo) * HF + col0 + (lane & 15);
#pragma unroll
  for (int r = 0; r < 8; ++r) dst[(size_t)r * HF] = acc[r];
}

// ---------------------------------------------------------------------------
// K2: per-node attention half-logits  s_src[n,h] = <h[n,h,:], a_src[h,:]>
// ---------------------------------------------------------------------------
__global__ void gat_scores(const float* __restrict__ Hbuf,
                           const float* __restrict__ a_src,
                           const float* __restrict__ a_dst,
                           float* __restrict__ s_src,
                           float* __restrict__ s_dst) {
  int idx = blockIdx.x * blockDim.x + threadIdx.x;
  if (idx >= N_NODES * HEADS) return;
  int n = idx / HEADS, hh = idx % HEADS;
  const float* hv = Hbuf + (size_t)n * HF + hh * OUT_F;
  const float* as = a_src + hh * OUT_F;
  const float* ad = a_dst + hh * OUT_F;
  float ss = 0.f, sd = 0.f;
#pragma unroll
  for (int f = 0; f < OUT_F; f += 4) {
    float4 h4 = *(const float4*)(hv + f);
    float4 s4 = *(const float4*)(as + f);
    float4 d4 = *(const float4*)(ad + f);
    ss += h4.x * s4.x + h4.y * s4.y + h4.z * s4.z + h4.w * s4.w;
    sd += h4.x * d4.x + h4.y * d4.y + h4.z * d4.z + h4.w * d4.w;
  }
  s_src[idx] = ss;
  s_dst[idx] = sd;
}

// ---------------------------------------------------------------------------
// K3: zero output, init segment max/sum
// ---------------------------------------------------------------------------
__global__ void gat_init(float* __restrict__ out, float* __restrict__ nmax,
                         float* __restrict__ nsum) {
  int idx = blockIdx.x * blockDim.x + threadIdx.x;
  if (idx < N_NODES * HF) out[idx] = 0.f;
  if (idx < N_NODES * HEADS) {
    nmax[idx] = -INFINITY;
    nsum[idx] = 0.f;
  }
}

__device__ __forceinline__ float leaky(float e) {
  return e > 0.f ? e : NEG_SLOPE * e;
}

// bit-trick float atomic max: signed-int max handles >=0, unsigned min handles <0
__device__ __forceinline__ void atomicMaxF(float* addr, float val) {
  if (!(val < 0.f))
    atomicMax((int*)addr, __float_as_int(val));
  else
    atomicMin((unsigned int*)addr, (unsigned int)__float_as_int(val));
}

// ---------------------------------------------------------------------------
// K4: segment max of edge logits over dst
// ---------------------------------------------------------------------------
__global__ void gat_edge_max(const int* __restrict__ ei,
                             const float* __restrict__ s_src,
                             const float* __restrict__ s_dst,
                             float* __restrict__ nmax) {
  int e = blockIdx.x * blockDim.x + threadIdx.x;
  if (e >= N_EDGES) return;
  int src = ei[e], dst = ei[N_EDGES + e];
#pragma unroll
  for (int hh = 0; hh < HEADS; ++hh) {
    float v = leaky(s_src[src * HEADS + hh] + s_dst[dst * HEADS + hh]);
    atomicMaxF(&nmax[dst * HEADS + hh], v);
  }
}

// K5: empty segments (max == -inf) -> 0, matching the reference
__global__ void gat_fix_max(float* __restrict__ nmax) {
  int idx = blockIdx.x * blockDim.x + threadIdx.x;
  if (idx >= N_NODES * HEADS) return;
  float m = nmax[idx];
  if (!isfinite(m)) nmax[idx] = 0.f;
}

// ---------------------------------------------------------------------------
// K6: segment sum of exp(e - max[dst])
// ---------------------------------------------------------------------------
__global__ void gat_edge_sum(const int* __restrict__ ei,
                             const float* __restrict__ s_src,
                             const float* __restrict__ s_dst,
                             const float* __restrict__ nmax,
                             float* __restrict__ nsum) {
  int e = blockIdx.x * blockDim.x + threadIdx.x;
  if (e >= N_EDGES) return;
  int src = ei[e], dst = ei[N_EDGES + e];
#pragma unroll
  for (int hh = 0; hh < HEADS; ++hh) {
    float v = leaky(s_src[src * HEADS + hh] + s_dst[dst * HEADS + hh]);
    float ex = __expf(v - nmax[dst * HEADS + hh]);
    atomicAdd(&nsum[dst * HEADS + hh], ex);
  }
}

// ---------------------------------------------------------------------------
// K7: out[dst] += attention * h[src]   (one thread per edge-head, 128-bit
// gathers of h[src], f32 L2 atomics for the scatter)
// ---------------------------------------------------------------------------
__global__ void gat_aggregate(const int* __restrict__ ei,
                              const float* __restrict__ s_src,
                              const float* __restrict__ s_dst,
                              const float* __restrict__ nmax,
                              const float* __restrict__ nsum,
                              const float* __restrict__ Hbuf,
                              float* __restrict__ out) {
  int idx = blockIdx.x * blockDim.x + threadIdx.x;
  if (idx >= N_EDGES * HEADS) return;
  int e = idx >> 2, hh = idx & 3;
  int src = ei[e], dst = ei[N_EDGES + e];
  float v = leaky(s_src[src * HEADS + hh] + s_dst[dst * HEADS + hh]);
  float ex = __expf(v - nmax[dst * HEADS + hh]);
  float att = ex / (nsum[dst * HEADS + hh] + GAT_EPS);
  const float* hv = Hbuf + (size_t)src * HF + hh * OUT_F;
  float* op = out + (size_t)dst * HF + hh * OUT_F;
#pragma unroll
  for (int f = 0; f < OUT_F; f += 4) {
    float4 h4 = *(const float4*)(hv + f);
    atomicAdd(op + f + 0, att * h4.x);
    atomicAdd(op + f + 1, att * h4.y);
    atomicAdd(op + f + 2, att * h4.z);
    atomicAdd(op + f + 3, att * h4.w);
  }
}

// ---------------------------------------------------------------------------
extern "C" void kernel_launch(void* const* d_in, const int* in_sizes, int n_in,
                              void* d_out, int out_size, void* d_ws, size_t ws_size,
                              hipStream_t stream) {
  const float* x      = (const float*)d_in[0];
  const int*   ei     = (const int*)d_in[1];
  const float* W      = (const float*)d_in[2];
  const float* a_src  = (const float*)d_in[3];
  const float* a_dst  = (const float*)d_in[4];
  float* out = (float*)d_out;

  // workspace layout (floats)
  float* ws    = (float*)d_ws;
  float* Hbuf  = ws;                                   // N*HF = 6,400,000
  float* s_src = Hbuf + (size_t)N_NODES * HF;          // N*H  =   200,000
  float* s_dst = s_src + (size_t)N_NODES * HEADS;
  float* nmax  = s_dst + (size_t)N_NODES * HEADS;
  float* nsum  = nmax + (size_t)N_NODES * HEADS;

  // K1: WMMA GEMM  (3125 row tiles x 8 col tiles; 4 waves/block, 2 y-blocks)
  gat_gemm_wmma<<<dim3(N_NODES / 16, 2, 1), 128, 0, stream>>>(x, W, Hbuf);

  // K2: per-node attention half-logits
  {
    int n = N_NODES * HEADS;
    gat_scores<<<(n + 255) / 256, 256, 0, stream>>>(Hbuf, a_src, a_dst, s_src, s_dst);
  }

  // K3: init out / max / sum
  {
    int n = N_NODES * HF;
    gat_init<<<(n + 255) / 256, 256, 0, stream>>>(out, nmax, nsum);
  }

  // K4: segment max
  gat_edge_max<<<(N_EDGES + 255) / 256, 256, 0, stream>>>(ei, s_src, s_dst, nmax);

  // K5: empty segments -> 0
  {
    int n = N_NODES * HEADS;
    gat_fix_max<<<(n + 255) / 256, 256, 0, stream>>>(nmax);
  }

  // K6: segment sum of exp
  gat_edge_sum<<<(N_EDGES + 255) / 256, 256, 0, stream>>>(ei, s_src, s_dst, nmax, nsum);

  // K7: weighted scatter-add aggregation
  {
    int n = N_EDGES * HEADS;
    gat_aggregate<<<(n + 255) / 256, 256, 0, stream>>>(ei, s_src, s_dst, nmax, nsum,
                                                       Hbuf, out);
  }
}